// PhysicsEnergyLoss_82532091560339
// MI455X (gfx1250) — compile-verified
//
#include <hip/hip_runtime.h>

typedef __attribute__((ext_vector_type(2))) float v2f;
typedef __attribute__((ext_vector_type(8))) float v8f;

#define BATCH 8
#define NPTS 2048
#define NT (NPTS / 16)                     // 128 tiles per dimension
#define TILES_PER_BATCH (NT * NT)          // 16384
#define TOTAL_TILES (BATCH * TILES_PER_BATCH) // 131072
#define TILES_PER_WAVE 32
#define WAVES_TOTAL (TOTAL_TILES / TILES_PER_WAVE) // 4096
#define WAVES_PER_BLOCK 8
#define BLOCK_THREADS (WAVES_PER_BLOCK * 32)       // 256
#define PAIR_BLOCKS (WAVES_TOTAL / WAVES_PER_BLOCK) // 512
#define PREP_BLOCKS ((BATCH * NPTS) / 256)          // 64

// Workspace layout (in floats)
#define WS_N2 0
#define WS_SUMA (BATCH * NPTS)              // 3 * PREP_BLOCKS floats
#define WS_DIST (WS_SUMA + 3 * PREP_BLOCKS) // PAIR_BLOCKS floats

// ---------------------------------------------------------------------------
// Kernel 1: per-point squared norms + partial sums of y, relu(r - y), r
// ---------------------------------------------------------------------------
__global__ void __launch_bounds__(256)
prep_kernel(const float* __restrict__ pos, const float* __restrict__ rad,
            float* __restrict__ ws) {
    const int idx = blockIdx.x * 256 + threadIdx.x; // < BATCH*NPTS by grid
    const float x = pos[idx * 3 + 0];
    const float y = pos[idx * 3 + 1];
    const float z = pos[idx * 3 + 2];
    const float r = rad[idx];
    ws[WS_N2 + idx] = x * x + y * y + z * z;

    float vy = y;
    float vg = fmaxf(r - y, 0.0f);
    float vr = r;
    #pragma unroll
    for (int off = 16; off > 0; off >>= 1) {
        vy += __shfl_down(vy, off, 32);
        vg += __shfl_down(vg, off, 32);
        vr += __shfl_down(vr, off, 32);
    }
    __shared__ float s[3][WAVES_PER_BLOCK];
    const int wave = threadIdx.x >> 5;
    const int lane = threadIdx.x & 31;
    if (lane == 0) { s[0][wave] = vy; s[1][wave] = vg; s[2][wave] = vr; }
    __syncthreads();
    if (threadIdx.x == 0) {
        float a = 0.f, b = 0.f, c = 0.f;
        #pragma unroll
        for (int w = 0; w < WAVES_PER_BLOCK; ++w) { a += s[0][w]; b += s[1][w]; c += s[2][w]; }
        ws[WS_SUMA + blockIdx.x * 3 + 0] = a;
        ws[WS_SUMA + blockIdx.x * 3 + 1] = b;
        ws[WS_SUMA + blockIdx.x * 3 + 2] = c;
    }
}

// ---------------------------------------------------------------------------
// Kernel 2: WMMA Gram-matrix pairwise distance sum.
//   dist(i,j) = sqrt(relu(n2[i] + n2[j] - 2 * dot(p_i, p_j)))
// One V_WMMA_F32_16X16X4_F32 per 16x16 tile (K = 3 padded to 4 with zeros).
// Epilogue uses raw v_sqrt_f32 (~1 ulp) to avoid the IEEE fixup sequence.
// ---------------------------------------------------------------------------
__global__ void __launch_bounds__(BLOCK_THREADS)
pair_kernel(const float* __restrict__ pos, const float* __restrict__ ws_n2,
            float* __restrict__ ws_dist) {
    const int lane      = threadIdx.x & 31;
    const int waveInBlk = threadIdx.x >> 5;
    const int wave      = blockIdx.x * WAVES_PER_BLOCK + waveInBlk;
    const int lo        = lane & 15;
    const int hi        = lane >> 4;

    // 32 consecutive tiles per wave; alignment guarantees constant (b, ti).
    const int t0   = wave * TILES_PER_WAVE;
    const int b    = t0 / TILES_PER_BATCH;
    const int rem0 = t0 % TILES_PER_BATCH;
    const int ti   = rem0 / NT;
    const int tj0  = rem0 % NT;

    const float* __restrict__ posb = pos + (size_t)b * NPTS * 3;
    const float* __restrict__ n2b  = ws_n2 + b * NPTS;

    // A fragment (16x4 f32): lanes 0-15 -> K0,K1 ; lanes 16-31 -> K2,pad0
    const int row = ti * 16 + lo;
    v2f afrag;
    if (hi == 0) {
        afrag.x = posb[row * 3 + 0];
        afrag.y = posb[row * 3 + 1];
    } else {
        afrag.x = posb[row * 3 + 2];
        afrag.y = 0.0f;
    }
    // Row norms for this lane's 8 output elements: M = v + 8*hi
    float ni[8];
    #pragma unroll
    for (int v = 0; v < 8; ++v) ni[v] = n2b[ti * 16 + v + 8 * hi];

    // Two independent accumulators to break the serial add chain.
    float acc0 = 0.0f, acc1 = 0.0f;
    for (int tt = 0; tt < TILES_PER_WAVE; ++tt) {
        const int col = (tj0 + tt) * 16 + lo;
        v2f bfrag;
        if (hi == 0) {
            bfrag.x = posb[col * 3 + 0];
            bfrag.y = posb[col * 3 + 1];
        } else {
            bfrag.x = posb[col * 3 + 2];
            bfrag.y = 0.0f;
        }
        v8f c = {};
        c = __builtin_amdgcn_wmma_f32_16x16x4_f32(
                /*neg_a=*/false, afrag, /*neg_b=*/false, bfrag,
                /*c_mod=*/(short)0, c, /*reuse_a=*/false, /*reuse_b=*/false);
        const float nj = n2b[col]; // N = lo, fixed per lane
        #pragma unroll
        for (int v = 0; v < 8; v += 2) {
            const float sq0 = ni[v + 0] + nj - 2.0f * c[v + 0];
            const float sq1 = ni[v + 1] + nj - 2.0f * c[v + 1];
            acc0 += __builtin_amdgcn_sqrtf(fmaxf(sq0, 0.0f));
            acc1 += __builtin_amdgcn_sqrtf(fmaxf(sq1, 0.0f));
        }
    }
    float acc = acc0 + acc1;

    #pragma unroll
    for (int off = 16; off > 0; off >>= 1) acc += __shfl_down(acc, off, 32);
    __shared__ float s[WAVES_PER_BLOCK];
    if (lane == 0) s[waveInBlk] = acc;
    __syncthreads();
    if (threadIdx.x == 0) {
        float t = 0.f;
        #pragma unroll
        for (int w = 0; w < WAVES_PER_BLOCK; ++w) t += s[w];
        ws_dist[blockIdx.x] = t;
    }
}

// ---------------------------------------------------------------------------
// Kernel 3: final reduction + loss assembly (single block)
// ---------------------------------------------------------------------------
__global__ void __launch_bounds__(256)
final_kernel(const float* __restrict__ ws, float* __restrict__ out) {
    float sy = 0.f, sg = 0.f, sr = 0.f, sd = 0.f;
    for (int i = threadIdx.x; i < PREP_BLOCKS; i += 256) {
        sy += ws[WS_SUMA + i * 3 + 0];
        sg += ws[WS_SUMA + i * 3 + 1];
        sr += ws[WS_SUMA + i * 3 + 2];
    }
    for (int i = threadIdx.x; i < PAIR_BLOCKS; i += 256) {
        sd += ws[WS_DIST + i];
    }
    #pragma unroll
    for (int off = 16; off > 0; off >>= 1) {
        sy += __shfl_down(sy, off, 32);
        sg += __shfl_down(sg, off, 32);
        sr += __shfl_down(sr, off, 32);
        sd += __shfl_down(sd, off, 32);
    }
    __shared__ float s[4][WAVES_PER_BLOCK];
    const int wave = threadIdx.x >> 5;
    const int lane = threadIdx.x & 31;
    if (lane == 0) { s[0][wave] = sy; s[1][wave] = sg; s[2][wave] = sr; s[3][wave] = sd; }
    __syncthreads();
    if (threadIdx.x == 0) {
        float ty = 0.f, tg = 0.f, tr = 0.f, td = 0.f;
        #pragma unroll
        for (int w = 0; w < WAVES_PER_BLOCK; ++w) {
            ty += s[0][w]; tg += s[1][w]; tr += s[2][w]; td += s[3][w];
        }
        const float invBN  = 1.0f / (float)(BATCH * NPTS);
        const float invBNN = 1.0f / ((float)BATCH * (float)NPTS * (float)NPTS);
        const float gravity = ty * invBN;
        const float ground  = tg * invBN;
        const float m = 2.0f * tr * invBN - td * invBNN; // mean(r_i + r_j - dist)
        // numerically stable softplus
        const float sp = (m > 0.0f) ? (m + log1pf(__expf(-m))) : log1pf(__expf(m));
        out[0] = 0.2f * gravity + 0.2f * ground + 0.2f * sp;
    }
}

// ---------------------------------------------------------------------------
extern "C" void kernel_launch(void* const* d_in, const int* in_sizes, int n_in,
                              void* d_out, int out_size, void* d_ws, size_t ws_size,
                              hipStream_t stream) {
    const float* pos = (const float*)d_in[0];  // [8, 2048, 3]
    const float* rad = (const float*)d_in[1];  // [8, 2048]
    float* ws  = (float*)d_ws;
    float* out = (float*)d_out;

    prep_kernel<<<PREP_BLOCKS, 256, 0, stream>>>(pos, rad, ws);
    pair_kernel<<<PAIR_BLOCKS, BLOCK_THREADS, 0, stream>>>(pos, ws + WS_N2, ws + WS_DIST);
    final_kernel<<<1, 256, 0, stream>>>(ws, out);
}